// UResNet_44590350467188
// MI455X (gfx1250) — compile-verified
//
#include <hip/hip_runtime.h>
#include <stdint.h>

// Problem constants (match reference)
#define SS      768
#define MCH     16
#define NCLS    5
#define BN_EPSf 0.0001f

typedef float v2f __attribute__((ext_vector_type(2)));
typedef float v8f __attribute__((ext_vector_type(8)));

// ---------------------------------------------------------------------------
// Kernel A: zero the occupancy bitmap (768^3 bits = 56.6 MB; fits in 192MB L2)
// ---------------------------------------------------------------------------
__global__ void k_zero(uint32_t* __restrict__ bm, int nwords) {
  int i = blockIdx.x * blockDim.x + threadIdx.x;
  int stride = gridDim.x * blockDim.x;
  for (; i < nwords; i += stride) bm[i] = 0u;
}

// ---------------------------------------------------------------------------
// Kernel B: set one bit per point at (x*S + y)*S + z
// ---------------------------------------------------------------------------
__global__ void k_build(const float* __restrict__ pc, uint32_t* __restrict__ bm, int n) {
  int i = blockIdx.x * blockDim.x + threadIdx.x;
  if (i >= n) return;
  int x = (int)pc[(size_t)i * 5 + 0];
  int y = (int)pc[(size_t)i * 5 + 1];
  int z = (int)pc[(size_t)i * 5 + 2];
  uint32_t idx = ((uint32_t)x * SS + (uint32_t)y) * SS + (uint32_t)z;
  atomicOr(&bm[idx >> 5], 1u << (idx & 31));
}

// ---------------------------------------------------------------------------
// Kernel C: per-point 27-bit neighbor hit mask (L2-resident bitmap probes),
// then out^T = W^T x H^T via 7 chained V_WMMA_F32_16X16X4_F32 (K=28, last
// slot zero-padded), then deterministic per-block sum / sumsq reduction.
// 256 threads = 8 wave32, each wave owns a 16-point tile.
// D layout: lane = point (n = lane&15), VGPR r = channel (c = r + 8*(lane>>4)).
// ---------------------------------------------------------------------------
__global__ void k_mask_stats(const float* __restrict__ pc,
                             const uint32_t* __restrict__ bm,
                             const float* __restrict__ W,
                             uint32_t* __restrict__ masks,
                             float* __restrict__ partials, int n) {
  const int lane = threadIdx.x & 31;
  const int wave = threadIdx.x >> 5;
  const int half = lane >> 4;
  const int lid  = lane & 15;
  const int base = blockIdx.x * 128 + wave * 16;
  const bool valid = (base + lid) < n;

  // lower half-wave computes the mask; upper half gets it via shuffle
  uint32_t mask = 0;
  if (half == 0 && valid) {
    const size_t pi = (size_t)(base + lid);
    int x = (int)pc[pi * 5 + 0];
    int y = (int)pc[pi * 5 + 1];
    int z = (int)pc[pi * 5 + 2];
#pragma unroll
    for (int k = 0; k < 27; ++k) {        // k order: dx outer, dy, dz inner (matches W[k])
      int dx = k / 9 - 1, dy = (k / 3) % 3 - 1, dz = k % 3 - 1;
      int nx = x + dx, ny = y + dy, nz = z + dz;
      if (nx >= 0 && nx < SS && ny >= 0 && ny < SS && nz >= 0 && nz < SS) {
        uint32_t idx = ((uint32_t)nx * SS + (uint32_t)ny) * SS + (uint32_t)nz;
        mask |= ((bm[idx >> 5] >> (idx & 31)) & 1u) << k;
      }
    }
    masks[base + lid] = mask;
  }
  mask = __shfl(mask, lid, 32);           // both half-waves hold point lid's mask

  // out^T[c][p] = sum_k W[k][c] * hit[p][k]
  v8f c = {};
#pragma unroll
  for (int t = 0; t < 7; ++t) {
    v2f a, b;
#pragma unroll
    for (int j = 0; j < 2; ++j) {
      int k = 4 * t + 2 * half + j;
      a[j] = (k < 27) ? W[k * MCH + lid] : 0.0f;                 // A = W^T (M=channel)
      b[j] = (k < 27) ? (float)((mask >> k) & 1u) : 0.0f;        // B = H^T (N=point)
    }
    c = __builtin_amdgcn_wmma_f32_16x16x4_f32(false, a, false, b, (short)0, c, false, false);
  }

  // per-channel sum & sumsq over this wave's 16 points (lanes within a half)
  __shared__ float lds_s[8 * 16];
  __shared__ float lds_q[8 * 16];
#pragma unroll
  for (int r = 0; r < 8; ++r) {
    float s = c[r], q = c[r] * c[r];
#pragma unroll
    for (int off = 1; off < 16; off <<= 1) {
      s += __shfl_xor(s, off, 32);
      q += __shfl_xor(q, off, 32);
    }
    if (lid == 0) {
      lds_s[wave * 16 + half * 8 + r] = s;
      lds_q[wave * 16 + half * 8 + r] = q;
    }
  }
  __syncthreads();
  if (threadIdx.x < 32) {
    int ch = threadIdx.x & 15;
    bool sq = threadIdx.x >= 16;
    const float* src = sq ? lds_q : lds_s;
    float t = 0.0f;
#pragma unroll
    for (int w = 0; w < 8; ++w) t += src[w * 16 + ch];
    partials[(size_t)blockIdx.x * 32 + (sq ? 16 : 0) + ch] = t;
  }
}

// ---------------------------------------------------------------------------
// Kernel D: fixed-order reduction of block partials -> BN scale/shift
// ---------------------------------------------------------------------------
__global__ void k_stats(const float* __restrict__ partials,
                        const float* __restrict__ gamma,
                        const float* __restrict__ beta,
                        float* __restrict__ stats, int nblocks, int n) {
  int c = threadIdx.x;
  if (c >= MCH) return;
  double s = 0.0, q = 0.0;
  for (int b = 0; b < nblocks; ++b) {
    s += (double)partials[(size_t)b * 32 + c];
    q += (double)partials[(size_t)b * 32 + 16 + c];
  }
  float mean = (float)(s / (double)n);
  float var  = (float)(q / (double)n) - mean * mean;
  float sc = gamma[c] * rsqrtf(var + BN_EPSf);
  stats[c]       = sc;
  stats[16 + c]  = beta[c] - mean * sc;
}

// ---------------------------------------------------------------------------
// Kernel E: recompute out tile from stored masks (WMMA #1), BN+ReLU in
// registers, then y^T = lin_w^T x h^T (WMMA #2, K=16 via 4 chained ops),
// write N x 5 logits. h already sits in the right lane=point layout; a
// single shfl_xor(.,16) exposes the other half's channels for B of WMMA #2.
// ---------------------------------------------------------------------------
__global__ void k_output(const uint32_t* __restrict__ masks,
                         const float* __restrict__ W,
                         const float* __restrict__ stats,
                         const float* __restrict__ lin_w,
                         const float* __restrict__ lin_b,
                         float* __restrict__ out, int n) {
  const int lane = threadIdx.x & 31;
  const int wave = threadIdx.x >> 5;
  const int half = lane >> 4;
  const int lid  = lane & 15;
  const int base = blockIdx.x * 128 + wave * 16;
  const bool valid = (base + lid) < n;

  uint32_t mask = valid ? masks[base + lid] : 0u;

  v8f c = {};
#pragma unroll
  for (int t = 0; t < 7; ++t) {
    v2f a, b;
#pragma unroll
    for (int j = 0; j < 2; ++j) {
      int k = 4 * t + 2 * half + j;
      a[j] = (k < 27) ? W[k * MCH + lid] : 0.0f;
      b[j] = (k < 27) ? (float)((mask >> k) & 1u) : 0.0f;
    }
    c = __builtin_amdgcn_wmma_f32_16x16x4_f32(false, a, false, b, (short)0, c, false, false);
  }

  // BN + ReLU: channel of c[r] is r + 8*half
  float h[8];
#pragma unroll
  for (int r = 0; r < 8; ++r) {
    int ch = r + 8 * half;
    float v = c[r] * stats[ch] + stats[16 + ch];
    h[r] = v > 0.0f ? v : 0.0f;
  }
  float hx[8];                       // same point's other 8 channels
#pragma unroll
  for (int r = 0; r < 8; ++r) hx[r] = __shfl_xor(h[r], 16, 32);

  // y^T[class][point] = sum_ch lin_w[ch][class] * h[point][ch]
  v8f y = {};
#pragma unroll
  for (int t = 0; t < 4; ++t) {
    v2f a, b;
#pragma unroll
    for (int j = 0; j < 2; ++j) {
      int k = 4 * t + 2 * half + j;                    // channel index 0..15
      a[j] = (lid < NCLS) ? lin_w[k * NCLS + lid] : 0.0f;   // A = lin_w^T (M=class)
      b[j] = ((k >> 3) == half) ? h[k & 7] : hx[k & 7];     // B = h^T (N=point)
    }
    y = __builtin_amdgcn_wmma_f32_16x16x4_f32(false, a, false, b, (short)0, y, false, false);
  }

  // y: lane = point, VGPR r -> class r (half 0 holds classes 0..7; only 0..4 real)
  if (half == 0 && valid) {
#pragma unroll
    for (int r = 0; r < NCLS; ++r)
      out[(size_t)(base + lid) * NCLS + r] = y[r] + lin_b[r];
  }
}

// ---------------------------------------------------------------------------
extern "C" void kernel_launch(void* const* d_in, const int* in_sizes, int n_in,
                              void* d_out, int out_size, void* d_ws, size_t ws_size,
                              hipStream_t stream) {
  const float* pc    = (const float*)d_in[0];
  const float* W     = (const float*)d_in[1];
  const float* gamma = (const float*)d_in[2];
  const float* beta  = (const float*)d_in[3];
  const float* lin_w = (const float*)d_in[4];
  const float* lin_b = (const float*)d_in[5];
  float* out = (float*)d_out;

  const int n = in_sizes[0] / 5;                      // 2,000,000
  const int BITMAP_WORDS = (SS * SS * SS) / 32;       // 14,155,776 words = 56.6 MB
  const int nblocks = (n + 127) / 128;                // 15,625 (exact)

  uint8_t* ws = (uint8_t*)d_ws;
  uint32_t* bitmap   = (uint32_t*)ws;
  uint32_t* masks    = (uint32_t*)(ws + (size_t)BITMAP_WORDS * 4);
  float*    partials = (float*)(ws + (size_t)BITMAP_WORDS * 4 + (size_t)n * 4);
  float*    stats    = (float*)((uint8_t*)partials + (size_t)nblocks * 32 * 4);

  k_zero<<<2048, 256, 0, stream>>>(bitmap, BITMAP_WORDS);
  k_build<<<(n + 255) / 256, 256, 0, stream>>>(pc, bitmap, n);
  k_mask_stats<<<nblocks, 256, 0, stream>>>(pc, bitmap, W, masks, partials, n);
  k_stats<<<1, 32, 0, stream>>>(partials, gamma, beta, stats, nblocks, n);
  k_output<<<nblocks, 256, 0, stream>>>(masks, W, stats, lin_w, lin_b, out, n);
}